// CrossLayer_42167988912746
// MI455X (gfx1250) — compile-verified
//
#include <hip/hip_runtime.h>
#include <hip/hip_bf16.h>

typedef float v2f __attribute__((ext_vector_type(2)));
typedef float v8f __attribute__((ext_vector_type(8)));

// ---------------------------------------------------------------------------
// Kernel 0: tiny precompute.
//   ws[0..2] = c_l = (sum_{j<l} b_j) . w_l  -> c0=0, c1=b0.w1, c2=(b0+b1).w2
//   ws[4..1027] = bsum[c] = b0+b1+b2
// ---------------------------------------------------------------------------
__global__ void dcn_pre_kernel(const float* __restrict__ W,
                               const float* __restrict__ b,
                               float* __restrict__ ws) {
  __shared__ float r1[1024];
  __shared__ float r2[1024];
  const int t = threadIdx.x;              // 1024 threads, t == column
  const float b0 = b[t];
  const float b1 = b[1024 + t];
  const float b2 = b[2048 + t];
  ws[4 + t] = b0 + b1 + b2;               // bsum
  r1[t] = b0 * W[1024 + t];               // partial of b0 . w1
  r2[t] = (b0 + b1) * W[2048 + t];        // partial of (b0+b1) . w2
  __syncthreads();
  for (int s = 512; s > 0; s >>= 1) {
    if (t < s) { r1[t] += r1[t + s]; r2[t] += r2[t + s]; }
    __syncthreads();
  }
  if (t == 0) { ws[0] = 0.0f; ws[1] = r1[0]; ws[2] = r2[0]; }
}

// ---------------------------------------------------------------------------
// Kernel 1: G = x0 @ W^T via V_WMMA_F32_16X16X4_F32 with 4-way split-K,
// then the per-row scalar recurrence.
//
// Block = 256 threads = 8 wave32 = 2 row-tiles x 4 K-slices.
// Each wave: A = 16x4 chunks of its tile's x rows (ISA 32-bit A layout:
// lanes 0-15 K-phase {0,1}, lanes 16-31 K-phase {2,3}); B = 4x16 with
// columns 0..2 = w0,w1,w2 staged in LDS. 64 accumulating WMMAs per wave
// over its K=256 slice; slice partials of D[r][n] = g_n(row r) are summed
// through LDS. alpha3 written to ws[1028 + row].
// ---------------------------------------------------------------------------
__global__ void dcn_gemv_wmma_kernel(const float* __restrict__ x,
                                     const float* __restrict__ W,
                                     const float* __restrict__ ws_c,
                                     float* __restrict__ alpha) {
  __shared__ float sW[3 * 1024];          // 12 KB: all three w rows
  __shared__ float sG[2 * 16 * 3 * 4];    // [tile][row][g_j][kslice] partials

  const int tid = threadIdx.x;            // blockDim.x == 256 (8 waves)
  for (int i = tid; i < 3 * 1024; i += 256) sW[i] = W[i];
  __syncthreads();

  const int wave  = tid >> 5;             // wave32
  const int lane  = tid & 31;
  const int tileI = wave >> 2;            // 0..1  : row-tile within block
  const int slice = wave & 3;             // 0..3  : K-slice
  const int half  = lane >> 4;            // 0: lanes 0-15 (K {0,1}); 1: lanes 16-31 (K {2,3})
  const int l16   = lane & 15;            // A: row M = l16 ; B: column N = l16
  const int row0  = (blockIdx.x * 2 + tileI) * 16;
  const int k0    = slice * 256;

  // A source: this lane's row base, offset by K-phase
  const float* arow = x + (size_t)(row0 + l16) * 1024 + half * 2;
  // B source: column n = l16 is w_n (valid only for n < 3), same K-phase
  const bool  bvalid = (l16 < 3);
  const float* brow = sW + l16 * 1024 + half * 2;
  const v2f bz = {0.0f, 0.0f};

  v8f acc = {};
  for (int k = k0; k < k0 + 256; k += 4) {
    v2f a = *(const v2f*)(arow + k);                  // global b64, 8B aligned
    v2f bm = bvalid ? *(const v2f*)(brow + k) : bz;   // ds b64 (or zero lanes)
    acc = __builtin_amdgcn_wmma_f32_16x16x4_f32(
        /*neg_a=*/false, a, /*neg_b=*/false, bm,
        /*c_mod=*/(short)0, acc, /*reuse_a=*/false, /*reuse_b=*/false);
  }

  // D layout: VGPR r, lanes 0-15 -> D[r][n]; lanes 16-31 -> D[r+8][n-16]
  if (bvalid) {
    #pragma unroll
    for (int r = 0; r < 8; ++r)
      sG[(((tileI * 16) + (r + 8 * half)) * 3 + l16) * 4 + slice] = acc[r];
  }
  __syncthreads();

  // One wave per tile (slice 0), lanes 0-15, finishes rows row0..row0+15.
  if (slice == 0 && lane < 16) {
    const int base = ((tileI * 16) + lane) * 3 * 4;
    const float g0 = sG[base + 0] + sG[base + 1] + sG[base + 2] + sG[base + 3];
    const float g1 = sG[base + 4] + sG[base + 5] + sG[base + 6] + sG[base + 7];
    const float g2 = sG[base + 8] + sG[base + 9] + sG[base + 10] + sG[base + 11];
    const float c1 = ws_c[1];
    const float c2 = ws_c[2];
    const float a1 = 1.0f + g0;
    const float a2 = a1 * (1.0f + g1) + c1;
    const float a3 = a2 * (1.0f + g2) + c2;
    alpha[row0 + lane] = a3;
  }
}

// ---------------------------------------------------------------------------
// Kernel 2: streaming epilogue, out = alpha3[row] * x0 + bsum[col].
// x0 re-read comes from the 192MB L2 (x fits easily). float4 coalesced b128.
// ---------------------------------------------------------------------------
__global__ void dcn_out_kernel(const float* __restrict__ x,
                               const float* __restrict__ bsum,
                               const float* __restrict__ alpha,
                               float* __restrict__ out) {
  const size_t gid  = (size_t)blockIdx.x * 256 + threadIdx.x;
  const size_t base = gid * 4;               // 4 floats per thread
  const int row = (int)(base >> 10);
  const int col = (int)(base & 1023);
  const float a = alpha[row];
  const float4 xv = *(const float4*)(x + base);
  const float4 bv = *(const float4*)(bsum + col);
  float4 o;
  o.x = a * xv.x + bv.x;
  o.y = a * xv.y + bv.y;
  o.z = a * xv.z + bv.z;
  o.w = a * xv.w + bv.w;
  *(float4*)(out + base) = o;
}

extern "C" void kernel_launch(void* const* d_in, const int* in_sizes, int n_in,
                              void* d_out, int out_size, void* d_ws, size_t ws_size,
                              hipStream_t stream) {
  (void)in_sizes; (void)n_in; (void)out_size; (void)ws_size;
  const float* x = (const float*)d_in[0];   // [8192, 1024]
  const float* W = (const float*)d_in[1];   // [3, 1024]
  const float* b = (const float*)d_in[2];   // [3, 1024]
  float* out = (float*)d_out;               // [8192, 1024]
  float* ws  = (float*)d_ws;                // [0..2]=c, [4..1027]=bsum, [1028..]=alpha

  float* ws_c     = ws;
  float* ws_bsum  = ws + 4;
  float* ws_alpha = ws + 1028;

  dcn_pre_kernel<<<1, 1024, 0, stream>>>(W, b, ws);
  // 8192 rows / (2 tiles * 16 rows per block) = 256 blocks of 256 threads,
  // 4-way split-K -> 2048 wave32 for the WMMA pass.
  dcn_gemv_wmma_kernel<<<256, 256, 0, stream>>>(x, W, ws_c, ws_alpha);
  // 8192*1024 floats / 4 per thread / 256 per block = 8192 blocks
  dcn_out_kernel<<<8192, 256, 0, stream>>>(x, ws_bsum, ws_alpha, out);
}